// Supermask_SVS2_AE_962072674553
// MI455X (gfx1250) — compile-verified
//
#include <hip/hip_runtime.h>
#include <hip/hip_bf16.h>

typedef float v2f __attribute__((ext_vector_type(2)));
typedef float v8f __attribute__((ext_vector_type(8)));

#define GS 0.5f   // GLOBAL_SPARSITY

// ---------------- Radix-select state ----------------
struct SelState {
  unsigned long long prefix;
  unsigned long long kthresh;
  long long remaining;
  int shift;
  int done;
  unsigned int hist[256];
};

__device__ __forceinline__ unsigned long long score_key(const float* s, long long i) {
  unsigned int ab = __float_as_uint(s[i]) & 0x7fffffffu;
  return ((unsigned long long)ab << 32) | (unsigned long long)(unsigned int)i;
}

__global__ void sel_init(SelState* st, const float* sr, int which, long long n) {
  float vs = sr[which];
  float t = tanhf(GS * vs * vs);
  float jf = floorf(t * (float)n);           // match jnp fp32 math
  long long j = (long long)jf;
  st->prefix = 0ull;
  st->shift = 56;
  st->done = 0;
  st->kthresh = 0ull;
  if (j <= 0) j = 0;
  if (j >= n) {                               // mask entirely zero
    st->done = 1;
    st->kthresh = ~0ull;
  }
  st->remaining = j;
  for (int i = 0; i < 256; ++i) st->hist[i] = 0u;
}

__global__ void sel_hist(const float* __restrict__ s, long long n, SelState* __restrict__ st) {
  __shared__ unsigned int h[256];
  int tid = threadIdx.x;
  h[tid] = 0u;
  __syncthreads();
  int shift = st->shift;
  unsigned long long prefix = st->prefix;
  unsigned long long himask = (shift >= 56) ? 0ull : (~0ull << (shift + 8));
  long long stride = (long long)gridDim.x * 256;
  for (long long i = (long long)blockIdx.x * 256 + tid; i < n; i += stride) {
    unsigned long long k = score_key(s, i);
    if ((k & himask) == prefix)
      atomicAdd(&h[(unsigned int)((k >> shift) & 0xFFull)], 1u);
  }
  __syncthreads();
  if (h[tid]) atomicAdd(&st->hist[tid], h[tid]);
}

__global__ void sel_scan(SelState* st) {
  if (!st->done) {
    long long rem = st->remaining;
    int shift = st->shift;
    int d;
    for (d = 0; d < 256; ++d) {
      long long c = (long long)st->hist[d];
      if (rem < c) break;
      rem -= c;
    }
    if (d > 255) d = 255;
    st->prefix |= ((unsigned long long)d) << shift;
    st->remaining = rem;
    st->shift = shift - 8;
    if (shift == 0) st->kthresh = st->prefix;
  }
  for (int i = 0; i < 256; ++i) st->hist[i] = 0u;
}

__global__ void apply_mask(float* __restrict__ wp, const float* __restrict__ w,
                           const float* __restrict__ s, const SelState* __restrict__ st,
                           long long n) {
  unsigned long long kth = st->kthresh;
  long long stride = (long long)gridDim.x * blockDim.x;
  for (long long i = (long long)blockIdx.x * blockDim.x + threadIdx.x; i < n; i += stride) {
    unsigned long long k = score_key(s, i);
    wp[i] = (k >= kth) ? w[i] : 0.0f;
  }
}

// ---------------- WMMA fp32 GEMM: H = X * W'^T + b (+ReLU) ----------------
// X: M x K row-major, W': N x K row-major, H: M x N row-major.
#define BM 128
#define BN 128
#define BK 16
#define LDA (BK + 4)   // padded LDS row stride (dwords): bank-conflict-free frag loads

__global__ void __launch_bounds__(256)
gemm_bias_act(float* __restrict__ C, const float* __restrict__ X,
              const float* __restrict__ W, const float* __restrict__ bias,
              int M, int N, int K, int relu) {
  __shared__ float As[2][BM * LDA];
  __shared__ float Bs[2][BN * LDA];

  const int tid  = threadIdx.x;
  const int lane = tid & 31;
  const int wave = tid >> 5;
  const int wm = (wave & 1) * 64;    // 2 waves along M, 64 rows each
  const int wn = (wave >> 1) * 32;   // 4 waves along N, 32 cols each
  const int bm = blockIdx.x * BM;
  const int bn = blockIdx.y * BN;
  const int l16 = lane & 15;
  const int lhi = lane >> 4;

  v8f acc[4][2];
#pragma unroll
  for (int i = 0; i < 4; ++i)
#pragma unroll
    for (int j = 0; j < 2; ++j) { v8f z{}; acc[i][j] = z; }

  float4 ar[2], br[2];

  auto loadRegs = [&](int kt) {
    const float* xp = X + (long long)bm * K + (long long)kt * BK;
    const float* wp = W + (long long)bn * K + (long long)kt * BK;
#pragma unroll
    for (int i = 0; i < 2; ++i) {
      int f   = tid + i * 256;   // float4 slot 0..511
      int row = f >> 2;          // 0..127
      int c4  = f & 3;           // 0..3
      ar[i] = *(const float4*)(xp + (long long)row * K + c4 * 4);
      br[i] = *(const float4*)(wp + (long long)row * K + c4 * 4);
    }
  };
  auto storeLds = [&](int buf) {
#pragma unroll
    for (int i = 0; i < 2; ++i) {
      int f   = tid + i * 256;
      int row = f >> 2;
      int c4  = f & 3;
      *(float4*)&As[buf][row * LDA + c4 * 4] = ar[i];
      *(float4*)&Bs[buf][row * LDA + c4 * 4] = br[i];
    }
  };
  auto compute = [&](int buf) {
#pragma unroll
    for (int kk = 0; kk < BK; kk += 4) {
      v2f a[4], b[2];
#pragma unroll
      for (int ms = 0; ms < 4; ++ms) {
        int m = wm + ms * 16 + l16;
        a[ms] = *(const v2f*)&As[buf][m * LDA + kk + lhi * 2];
      }
#pragma unroll
      for (int ns = 0; ns < 2; ++ns) {
        int nn = wn + ns * 16 + l16;
        b[ns] = *(const v2f*)&Bs[buf][nn * LDA + kk + lhi * 2];
      }
#pragma unroll
      for (int ms = 0; ms < 4; ++ms)
#pragma unroll
        for (int ns = 0; ns < 2; ++ns)
          acc[ms][ns] = __builtin_amdgcn_wmma_f32_16x16x4_f32(
              false, a[ms], false, b[ns], (short)0, acc[ms][ns], false, false);
    }
  };

  const int nT = K / BK;
  loadRegs(0);
  storeLds(0);
  __syncthreads();
  for (int kt = 0; kt < nT; ++kt) {
    int cur = kt & 1;
    if (kt + 1 < nT) loadRegs(kt + 1);
    compute(cur);
    __syncthreads();
    if (kt + 1 < nT) { storeLds(1 - cur); __syncthreads(); }
  }

  // Epilogue: C layout -> VGPR v: lanes0-15 row v, lanes16-31 row v+8
#pragma unroll
  for (int ms = 0; ms < 4; ++ms) {
#pragma unroll
    for (int ns = 0; ns < 2; ++ns) {
      int nn = bn + wn + ns * 16 + l16;
      float bv = bias[nn];
      int mbase = bm + wm + ms * 16 + lhi * 8;
      float* cp = C + (long long)mbase * N + nn;
#pragma unroll
      for (int v = 0; v < 8; ++v) {
        float val = acc[ms][ns][v] + bv;
        if (relu) val = fmaxf(val, 0.0f);
        cp[(long long)v * N] = val;
      }
    }
  }
}

// ---------------- Host side ----------------
extern "C" void kernel_launch(void* const* d_in, const int* in_sizes, int n_in,
                              void* d_out, int out_size, void* d_ws, size_t ws_size,
                              hipStream_t stream) {
  (void)in_sizes; (void)n_in; (void)out_size; (void)ws_size;
  static const int OUT[8] = {2048, 1024, 512, 256, 512, 1024, 2048, 4096};
  static const int IN[8]  = {4096, 2048, 1024, 512, 256, 512, 1024, 2048};
  const int M = 8192;

  const float* x  = (const float*)d_in[0];
  const float* sr = (const float*)d_in[1];

  float* base = (float*)d_ws;
  size_t off = 0;
  float* wp[8];
  for (int i = 0; i < 8; ++i) { wp[i] = base + off; off += (size_t)OUT[i] * IN[i]; }
  float* actA = base + off; off += (size_t)M * 2048;
  float* actB = base + off; off += (size_t)M * 2048;
  SelState* st = (SelState*)(base + off);

  const float* src = x;
  for (int i = 0; i < 8; ++i) {
    const float* w = (const float*)d_in[2 + 3 * i];
    const float* b = (const float*)d_in[3 + 3 * i];
    const float* s = (const float*)d_in[4 + 3 * i];
    long long n = (long long)OUT[i] * IN[i];

    sel_init<<<1, 1, 0, stream>>>(st, sr, i, n);
    int hb = (int)((n + 256 * 32 - 1) / (256 * 32));
    if (hb > 1024) hb = 1024;
    for (int p = 0; p < 8; ++p) {
      sel_hist<<<hb, 256, 0, stream>>>(s, n, st);
      sel_scan<<<1, 1, 0, stream>>>(st);
    }
    int mb = (int)((n + 1023) / 1024);
    if (mb > 2048) mb = 2048;
    apply_mask<<<mb, 1024, 0, stream>>>(wp[i], w, s, st, n);

    float* dst = (i == 7) ? (float*)d_out : ((i & 1) ? actB : actA);
    int relu = (i == 3 || i == 7) ? 0 : 1;
    dim3 grid(M / BM, OUT[i] / BN);
    gemm_bias_act<<<grid, 256, 0, stream>>>(dst, src, wp[i], b, M, OUT[i], IN[i], relu);
    src = dst;
  }
}